// MultiLoRALayerMaskingHom_54623394070735
// MI455X (gfx1250) — compile-verified
//
#include <hip/hip_runtime.h>

typedef __attribute__((ext_vector_type(2))) float v2f;
typedef __attribute__((ext_vector_type(8))) float v8f;

#define RANK       16
#define IN_F       4096
#define OUT_F      4096
#define BATCH      4
#define SEQ        2048
#define SCALING    1.0f      // ALPHA / RANK = 16/16
#define TILE_M     16
#define WAVES      8
#define THREADS    256       // 8 wave32 waves

// Fused multi-LoRA: per workgroup, one 16-row tile of x.
// Stage 1: Bx[16x16] = x_tile[16x4096] * B_sel^T  (K split over 8 waves)
// Stage 2: out_tile[16x4096] = Bx[16x16] * A_sel^T (N split over 8 waves)
__global__ __launch_bounds__(THREADS)
void multilora_fused_wmma(const float* __restrict__ x,
                          const float* __restrict__ A,
                          const float* __restrict__ Bw,
                          const int*   __restrict__ adapter_ids,
                          float*       __restrict__ out)
{
    __shared__ float lds_part[WAVES * 256]; // per-wave partial Bx (16x16 each)
    __shared__ float lds_bx[256];           // reduced Bx, row-major [m][r]

    const int tilesPerBatch = SEQ / TILE_M;        // 128
    const int wg   = blockIdx.x;
    const int b    = wg / tilesPerBatch;
    const int row0 = (wg % tilesPerBatch) * TILE_M;

    const int tid  = threadIdx.x;
    const int wave = tid >> 5;
    const int lane = tid & 31;
    const int half = lane >> 4;     // 0: K={0,1}, 1: K={2,3}
    const int l15  = lane & 15;     // M index (A frag) / N index (B frag)
    const int koff = half * 2;

    const int aid = adapter_ids[b];
    const float* __restrict__ Bm = Bw + (size_t)aid * RANK * IN_F;    // [16][4096]
    const float* __restrict__ Am = A  + (size_t)aid * OUT_F * RANK;   // [4096][16]
    const float* __restrict__ xt = x  + ((size_t)b * SEQ + row0) * IN_F;

    // -------- Stage 1 --------
    const int   kChunk = IN_F / WAVES;   // 512 -> 128 wmma steps per wave
    const int   k0     = wave * kChunk;
    const float* xrow  = xt + (size_t)l15 * IN_F + koff;  // A-frag base (row m)
    const float* brow  = Bm + (size_t)l15 * IN_F + koff;  // B-frag base (rank n)

    v8f acc0 = {}; v8f acc1 = {}; v8f acc2 = {}; v8f acc3 = {};
    for (int k = k0; k < k0 + kChunk; k += 16) {
        v2f a0 = *(const v2f*)(xrow + k);
        v2f w0 = *(const v2f*)(brow + k);
        v2f a1 = *(const v2f*)(xrow + k + 4);
        v2f w1 = *(const v2f*)(brow + k + 4);
        v2f a2 = *(const v2f*)(xrow + k + 8);
        v2f w2 = *(const v2f*)(brow + k + 8);
        v2f a3 = *(const v2f*)(xrow + k + 12);
        v2f w3 = *(const v2f*)(brow + k + 12);
        acc0 = __builtin_amdgcn_wmma_f32_16x16x4_f32(false, a0, false, w0, (short)0, acc0, false, false);
        acc1 = __builtin_amdgcn_wmma_f32_16x16x4_f32(false, a1, false, w1, (short)0, acc1, false, false);
        acc2 = __builtin_amdgcn_wmma_f32_16x16x4_f32(false, a2, false, w2, (short)0, acc2, false, false);
        acc3 = __builtin_amdgcn_wmma_f32_16x16x4_f32(false, a3, false, w3, (short)0, acc3, false, false);
    }
    v8f acc = (acc0 + acc1) + (acc2 + acc3);

    // C/D layout: VGPR v, lane -> element (M = v + half*8, N = l15)
    #pragma unroll
    for (int v = 0; v < 8; ++v)
        lds_part[wave * 256 + (v + half * 8) * 16 + l15] = acc[v];
    __syncthreads();

    // Deterministic cross-wave reduction (fixed order, no float atomics)
    {
        float s = 0.0f;
        #pragma unroll
        for (int w = 0; w < WAVES; ++w) s += lds_part[w * 256 + tid];
        lds_bx[tid] = s;
    }
    __syncthreads();

    // -------- Stage 2 --------
    // Preload Bx as four 16x4 A-fragments (K = 16 total)
    v2f af[4];
    #pragma unroll
    for (int j = 0; j < 4; ++j)
        af[j] = *(const v2f*)(&lds_bx[l15 * 16 + 4 * j + koff]);

    const int colChunk = OUT_F / WAVES;  // 512 -> 32 blocks of 16 cols
    const int col0w    = wave * colChunk;
    float* outt = out + ((size_t)b * SEQ + row0) * OUT_F;

    for (int blk = 0; blk < colChunk / 16; ++blk) {
        const int col0 = col0w + blk * 16;
        // B-frag: element (K=4j+koff+{0,1}, N=l15) = Am[col0+l15][4j+koff+{0,1}]
        const float* arow = Am + (size_t)(col0 + l15) * RANK + koff;
        v2f w0 = *(const v2f*)(arow);
        v2f w1 = *(const v2f*)(arow + 4);
        v2f w2 = *(const v2f*)(arow + 8);
        v2f w3 = *(const v2f*)(arow + 12);

        v8f o = {};
        o = __builtin_amdgcn_wmma_f32_16x16x4_f32(false, af[0], false, w0, (short)0, o, false, false);
        o = __builtin_amdgcn_wmma_f32_16x16x4_f32(false, af[1], false, w1, (short)0, o, false, false);
        o = __builtin_amdgcn_wmma_f32_16x16x4_f32(false, af[2], false, w2, (short)0, o, false, false);
        o = __builtin_amdgcn_wmma_f32_16x16x4_f32(false, af[3], false, w3, (short)0, o, false, false);

        #pragma unroll
        for (int v = 0; v < 8; ++v)
            outt[(size_t)(v + half * 8) * OUT_F + col0 + l15] = o[v] * SCALING;
    }
}

extern "C" void kernel_launch(void* const* d_in, const int* in_sizes, int n_in,
                              void* d_out, int out_size, void* d_ws, size_t ws_size,
                              hipStream_t stream) {
    (void)in_sizes; (void)n_in; (void)out_size; (void)d_ws; (void)ws_size;
    const float* x   = (const float*)d_in[0];
    const float* A   = (const float*)d_in[1];
    const float* B   = (const float*)d_in[2];
    const int*   ids = (const int*)d_in[3];
    float* out = (float*)d_out;

    dim3 grid(BATCH * (SEQ / TILE_M));   // 512 workgroups
    dim3 block(THREADS);                 // 256 threads = 8 wave32 waves
    hipLaunchKernelGGL(multilora_fused_wmma, grid, block, 0, stream,
                       x, A, B, ids, out);
}